// BGMSA_6227702579812
// MI455X (gfx1250) — compile-verified
//
#include <hip/hip_runtime.h>
#include <math.h>

typedef __attribute__((ext_vector_type(2))) float v2f;
typedef __attribute__((ext_vector_type(4))) float v4f;
typedef __attribute__((ext_vector_type(8))) float v8f;

#define BATCH 4
#define HGT 128
#define WID 128
#define CH 512
#define NTOK 16384      /* 128*128 */
#define MROWS 65536     /* BATCH*NTOK */
#define HEADS 8
#define DHEAD 64

__device__ __forceinline__ v8f wmma4(v2f a, v2f b, v8f c) {
    // V_WMMA_F32_16X16X4_F32 : D = A(16x4) * B(4x16) + C(16x16), f32 exact
    return __builtin_amdgcn_wmma_f32_16x16x4_f32(
        /*neg_a=*/false, a, /*neg_b=*/false, b,
        /*c_mod=*/(short)0, c, /*reuse_a=*/false, /*reuse_b=*/false);
}

// ---------------------------------------------------------------------------
// f32 GEMM: C[M,N] = A[M,K] @ B[K,N] (+bias). Row-major.
// Block = 128(M) x 64(N); 8 waves, each 16(M) x 64(N).
// B staged per 64-deep K-chunk into LDS, transposed [col][k] (stride 66:
// even for 8B-aligned ds_load_b64, spreads the 64 banks). Shared by all waves.
// Requires M % 128 == 0, N % 64 == 0, K % 64 == 0.
// ---------------------------------------------------------------------------
__global__ __launch_bounds__(256)
void gemm_f32_wmma(const float* __restrict__ A, const float* __restrict__ B,
                   const float* __restrict__ bias, float* __restrict__ C,
                   int M, int N, int K) {
    __shared__ __align__(16) float ldsBT[64 * 66];

    const int tid   = threadIdx.x;
    const int lane  = tid & 31;
    const int wave  = tid >> 5;                  // 0..7
    const int tilesN = N >> 6;
    const int mBlk  = blockIdx.x / tilesN;
    const int nBlk  = blockIdx.x - mBlk * tilesN;
    const int m0    = mBlk * 128 + wave * 16;
    const int n0    = nBlk << 6;
    const int mLane = lane & 15;
    const int kHalf = (lane >> 4) << 1;          // 0 or 2

    const float* aPtr = A + (size_t)(m0 + mLane) * K + kHalf;

    // cooperative B-chunk loader mapping: row = tid>>2 (0..63), 4 float4s/thread
    const int br = tid >> 2;
    const int bf = tid & 3;

    v8f acc0 = {}, acc1 = {}, acc2 = {}, acc3 = {};
    for (int kc = 0; kc < K; kc += 64) {
        __syncthreads();
        // load B[kc+br][n0 + 4*(bf+4i) .. +3] and scatter transposed into LDS
#pragma unroll
        for (int i = 0; i < 4; ++i) {
            const int f  = bf + (i << 2);        // float4 index 0..15
            const int c0 = f << 2;               // col 0..60
            v4f v = *(const v4f*)(B + (size_t)(kc + br) * N + n0 + c0);
            ldsBT[(c0 + 0) * 66 + br] = v.x;
            ldsBT[(c0 + 1) * 66 + br] = v.y;
            ldsBT[(c0 + 2) * 66 + br] = v.z;
            ldsBT[(c0 + 3) * 66 + br] = v.w;
        }
        __syncthreads();
#pragma unroll 4
        for (int kk = 0; kk < 64; kk += 4) {
            v2f a = *(const v2f*)aPtr;           // A[m][kc+kk+kHalf .. +1]
            aPtr += 4;
            v2f b0 = *(const v2f*)&ldsBT[(mLane +  0) * 66 + kk + kHalf];
            v2f b1 = *(const v2f*)&ldsBT[(mLane + 16) * 66 + kk + kHalf];
            v2f b2 = *(const v2f*)&ldsBT[(mLane + 32) * 66 + kk + kHalf];
            v2f b3 = *(const v2f*)&ldsBT[(mLane + 48) * 66 + kk + kHalf];
            acc0 = wmma4(a, b0, acc0);
            acc1 = wmma4(a, b1, acc1);
            acc2 = wmma4(a, b2, acc2);
            acc3 = wmma4(a, b3, acc3);
        }
    }

    float bv0 = 0.f, bv1 = 0.f, bv2 = 0.f, bv3 = 0.f;
    if (bias) {
        bv0 = bias[n0 + mLane];      bv1 = bias[n0 + 16 + mLane];
        bv2 = bias[n0 + 32 + mLane]; bv3 = bias[n0 + 48 + mLane];
    }
    const int mOff = (lane >> 4) << 3;           // rows 0..7 or 8..15
#pragma unroll
    for (int r = 0; r < 8; ++r) {
        float* out = C + (size_t)(m0 + mOff + r) * N + n0 + mLane;
        out[0]  = acc0[r] + bv0;
        out[16] = acc1[r] + bv1;
        out[32] = acc2[r] + bv2;
        out[48] = acc3[r] + bv3;
    }
}

// ---------------------------------------------------------------------------
// Column sum-of-squares over the n axis (for L2 norms of q,k rows).
// Stage 1: 64 chunks of 256 rows -> partials [src][b][chunk][512]
// ---------------------------------------------------------------------------
__global__ __launch_bounds__(256)
void colssq_partial(const float* __restrict__ q, const float* __restrict__ k,
                    float* __restrict__ part) {
    const int chunk = blockIdx.x;   // 0..63
    const int b     = blockIdx.y;   // 0..3
    const int src   = blockIdx.z;   // 0=q 1=k
    const float* A  = src ? k : q;
    const int c     = threadIdx.x << 1;
    const float* p  = A + ((size_t)b * NTOK + (size_t)chunk * 256) * CH + c;
    v2f s = {};
    for (int r = 0; r < 256; ++r) {
        v2f v = *(const v2f*)(p + (size_t)r * CH);
        s += v * v;
    }
    float* o = part + ((((size_t)src * BATCH + b) * 64) + chunk) * CH + c;
    o[0] = s.x; o[1] = s.y;
}

__global__ __launch_bounds__(256)
void colssq_reduce(const float* __restrict__ part,
                   float* __restrict__ n2q, float* __restrict__ n2k) {
    const int b   = blockIdx.x & 3;
    const int src = blockIdx.x >> 2;
    const int c   = threadIdx.x << 1;
    const float* p = part + (((size_t)src * BATCH + b) * 64) * CH + c;
    v2f s = {};
    for (int ch = 0; ch < 64; ++ch) s += *(const v2f*)(p + (size_t)ch * CH);
    float* o = (src ? n2k : n2q) + b * CH + c;
    o[0] = s.x; o[1] = s.y;
}

// ---------------------------------------------------------------------------
// attn_raw[b,h,d,e] = sum_n k[b,n,h*64+d] * q[b,n,h*64+e]
// One block per (b,h): 16 waves, each owns one 16x16 tile; K-dim n=16384
// processed in 32-row chunks staged into LDS (both k and q column slices),
// transposed [col][k] with stride 34 (even -> aligned b64, bank-spread).
// ---------------------------------------------------------------------------
__global__ __launch_bounds__(512)
void attn_kq(const float* __restrict__ kmat, const float* __restrict__ qmat,
             float* __restrict__ attn) {
    __shared__ __align__(16) float ldsK[64 * 34];
    __shared__ __align__(16) float ldsQ[64 * 34];

    const int bh   = blockIdx.x;                 // 0..31
    const int b    = bh >> 3, h = bh & 7;
    const int tid  = threadIdx.x;
    const int lane = tid & 31;
    const int wave = tid >> 5;                   // 0..15
    const int d0   = (wave >> 2) << 4;
    const int e0   = (wave & 3) << 4;
    const int kHalf = (lane >> 4) << 1;
    const int mLane = lane & 15;
    const int cbase = h << 6;
    const size_t rowBase = (size_t)b * NTOK;

    // loader mapping: waves 0-7 stage kmat, waves 8-15 stage qmat (wave-uniform)
    const int mat = tid >> 8;                    // 0=k 1=q
    const int t2  = tid & 255;
    const int lrow = t2 >> 3;                    // 0..31
    const int lf   = t2 & 7;                     // float4 index base
    const float* src = mat ? qmat : kmat;
    float* dstL      = mat ? ldsQ : ldsK;

    v8f acc = {};
    for (int n0v = 0; n0v < NTOK; n0v += 32) {
        __syncthreads();
#pragma unroll
        for (int i = 0; i < 2; ++i) {
            const int f  = lf + (i << 3);        // 0..15
            const int c0 = f << 2;
            v4f v = *(const v4f*)(src + (rowBase + n0v + lrow) * CH + cbase + c0);
            dstL[(c0 + 0) * 34 + lrow] = v.x;
            dstL[(c0 + 1) * 34 + lrow] = v.y;
            dstL[(c0 + 2) * 34 + lrow] = v.z;
            dstL[(c0 + 3) * 34 + lrow] = v.w;
        }
        __syncthreads();
#pragma unroll
        for (int kk = 0; kk < 32; kk += 4) {
            v2f a  = *(const v2f*)&ldsK[(d0 + mLane) * 34 + kk + kHalf];
            v2f bf = *(const v2f*)&ldsQ[(e0 + mLane) * 34 + kk + kHalf];
            acc = wmma4(a, bf, acc);
        }
    }
    const int mOff = (lane >> 4) << 3;
    float* out = attn + (size_t)bh * (DHEAD * DHEAD);
#pragma unroll
    for (int r = 0; r < 8; ++r)
        out[(d0 + mOff + r) * DHEAD + e0 + mLane] = acc[r];
}

// ---------------------------------------------------------------------------
// Fold L2 norms + rescale into logits, softmax over e (row length 64).
// One thread per row (b,h,d).
// ---------------------------------------------------------------------------
__global__ __launch_bounds__(256)
void attn_softmax(float* __restrict__ attn, const float* __restrict__ n2k,
                  const float* __restrict__ n2q, const float* __restrict__ rescale) {
    const int row = blockIdx.x * blockDim.x + threadIdx.x;   // 0..2047
    if (row >= 32 * DHEAD) return;
    const int d  = row & 63;
    const int bh = row >> 6;
    const int b  = bh >> 3, h = bh & 7;
    const int cb = h << 6;
    const float kn = fmaxf(sqrtf(n2k[b * CH + cb + d]), 1e-12f);
    const float rs = rescale[h];
    float* p = attn + (size_t)row * DHEAD;

    float vals[DHEAD];
    float mx = -INFINITY;
#pragma unroll 8
    for (int e = 0; e < DHEAD; ++e) {
        const float qn = fmaxf(sqrtf(n2q[b * CH + cb + e]), 1e-12f);
        const float v  = p[e] / (kn * qn) * rs;
        vals[e] = v;
        mx = fmaxf(mx, v);
    }
    float s = 0.f;
#pragma unroll 8
    for (int e = 0; e < DHEAD; ++e) { vals[e] = expf(vals[e] - mx); s += vals[e]; }
    const float inv = 1.0f / s;
#pragma unroll 8
    for (int e = 0; e < DHEAD; ++e) p[e] = vals[e] * inv;
}

// ---------------------------------------------------------------------------
// xo[b,n,h*64+d] = sum_e attn[b,h,d,e] * (v[b,n,h*64+e] * mask[b,n,h*64+e])
// GEMM per (b,h): M=n, N=64 (d), K=64 (e). One wave -> 16x64 strip.
// All fragment loads are contiguous 8B (attn table is L2-resident, 512KB).
// ---------------------------------------------------------------------------
__global__ __launch_bounds__(256)
void attn_apply_v(const float* __restrict__ vmat, const float* __restrict__ mask,
                  const float* __restrict__ attn, float* __restrict__ xo) {
    const int lane = threadIdx.x & 31;
    const int wave = threadIdx.x >> 5;
    const int tile = blockIdx.x * (blockDim.x >> 5) + wave;   // 0..32767
    const int tilesPerBH = NTOK >> 4;                         // 1024
    const int bh    = tile / tilesPerBH;
    const int mTile = tile - bh * tilesPerBH;
    const int b = bh >> 3, h = bh & 7;
    const int cbase = h << 6;
    const int mLane = lane & 15;
    const int kHalf = (lane >> 4) << 1;

    const size_t rowBase = (size_t)b * NTOK + (size_t)mTile * 16;
    const size_t aOff    = (rowBase + mLane) * CH + cbase + kHalf;
    const float* attnBase = attn + (size_t)bh * (DHEAD * DHEAD);

    v8f acc0 = {}, acc1 = {}, acc2 = {}, acc3 = {};
#pragma unroll 4
    for (int k = 0; k < DHEAD; k += 4) {
        v2f av = *(const v2f*)(vmat + aOff + k);
        v2f am = *(const v2f*)(mask + aOff + k);
        v2f a  = av * am;
        // B[kk][d] = attn[d][k+kk] : contiguous pair over e
        v2f b0 = *(const v2f*)(attnBase + (0  + mLane) * DHEAD + k + kHalf);
        v2f b1 = *(const v2f*)(attnBase + (16 + mLane) * DHEAD + k + kHalf);
        v2f b2 = *(const v2f*)(attnBase + (32 + mLane) * DHEAD + k + kHalf);
        v2f b3 = *(const v2f*)(attnBase + (48 + mLane) * DHEAD + k + kHalf);
        acc0 = wmma4(a, b0, acc0);
        acc1 = wmma4(a, b1, acc1);
        acc2 = wmma4(a, b2, acc2);
        acc3 = wmma4(a, b3, acc3);
    }
    const int mOff = (lane >> 4) << 3;
#pragma unroll
    for (int r = 0; r < 8; ++r) {
        float* out = xo + (rowBase + mOff + r) * CH + cbase + mLane;
        out[0]  = acc0[r];
        out[16] = acc1[r];
        out[32] = acc2[r];
        out[48] = acc3[r];
    }
}

// ---------------------------------------------------------------------------
// Depthwise 3x3 convs (NHWC, SAME, no bias). conv1 fused with tanh-GELU,
// conv2 accumulates into d_out (which already holds out_c).
// ---------------------------------------------------------------------------
__device__ __forceinline__ float gelu_tanh(float v) {
    return 0.5f * v * (1.0f + tanhf(0.7978845608028654f * (v + 0.044715f * v * v * v)));
}

__global__ __launch_bounds__(256)
void dwconv3x3(const float* __restrict__ in, const float* __restrict__ w,
               float* __restrict__ out, int doGelu, int accumulate) {
    const int idx = blockIdx.x;                  // b*H*W + y*W + x
    const int x = idx % WID;
    const int y = (idx / WID) % HGT;
    const int b = idx / (WID * HGT);
    const int c = threadIdx.x << 1;

    float s0 = 0.f, s1 = 0.f;
#pragma unroll
    for (int ky = -1; ky <= 1; ++ky) {
        const int yy = y + ky;
        if (yy < 0 || yy >= HGT) continue;
#pragma unroll
        for (int kx = -1; kx <= 1; ++kx) {
            const int xx = x + kx;
            if (xx < 0 || xx >= WID) continue;
            const float* p = in + (((size_t)b * HGT + yy) * WID + xx) * CH + c;
            const int wi = (ky + 1) * 3 + (kx + 1);
            s0 += p[0] * w[(size_t)c * 9 + wi];
            s1 += p[1] * w[(size_t)(c + 1) * 9 + wi];
        }
    }
    if (doGelu) { s0 = gelu_tanh(s0); s1 = gelu_tanh(s1); }
    float* o = out + (size_t)idx * CH + c;
    if (accumulate) { o[0] += s0; o[1] += s1; }
    else            { o[0]  = s0; o[1]  = s1; }
}

// ---------------------------------------------------------------------------
extern "C" void kernel_launch(void* const* d_in, const int* in_sizes, int n_in,
                              void* d_out, int out_size, void* d_ws, size_t ws_size,
                              hipStream_t stream) {
    const float* x_in    = (const float*)d_in[0];
    const float* mask    = (const float*)d_in[1];
    const float* Wq      = (const float*)d_in[2];
    const float* Wk      = (const float*)d_in[3];
    const float* Wv      = (const float*)d_in[4];
    const float* rescale = (const float*)d_in[5];
    const float* Wp      = (const float*)d_in[6];
    const float* bp      = (const float*)d_in[7];
    const float* c1w     = (const float*)d_in[8];
    const float* c2w     = (const float*)d_in[9];
    float* out = (float*)d_out;

    const size_t SZ = (size_t)MROWS * CH;        // 33.5M floats per tensor
    float* q_inp = (float*)d_ws;                 // later reused as xo
    float* k_inp = q_inp + SZ;                   // later reused as conv tmp
    float* v_inp = k_inp + SZ;
    float* attn  = v_inp + SZ;                   // 32*64*64
    float* n2q   = attn + (size_t)32 * DHEAD * DHEAD;
    float* n2k   = n2q + BATCH * CH;
    float* part  = n2k + BATCH * CH;             // 2*4*64*512

    const int gemmBlocks = (MROWS / 128) * (CH / 64);      // 4096

    // 1) Q/K/V projections (WMMA f32, LDS-staged weights)
    gemm_f32_wmma<<<gemmBlocks, 256, 0, stream>>>(x_in, Wq, nullptr, q_inp, MROWS, CH, CH);
    gemm_f32_wmma<<<gemmBlocks, 256, 0, stream>>>(x_in, Wk, nullptr, k_inp, MROWS, CH, CH);
    gemm_f32_wmma<<<gemmBlocks, 256, 0, stream>>>(x_in, Wv, nullptr, v_inp, MROWS, CH, CH);

    // 2) column L2 norms of q,k over n (two-stage, deterministic)
    colssq_partial<<<dim3(64, BATCH, 2), 256, 0, stream>>>(q_inp, k_inp, part);
    colssq_reduce<<<8, 256, 0, stream>>>(part, n2q, n2k);

    // 3) logits K·Qᵀ (WMMA, K-dim = 16384, LDS-staged chunks)
    attn_kq<<<32, 512, 0, stream>>>(k_inp, q_inp, attn);

    // 4) fold norms + rescale, softmax over e
    attn_softmax<<<(32 * DHEAD + 255) / 256, 256, 0, stream>>>(attn, n2k, n2q, rescale);

    // 5) positional branch stage 1: tmp = gelu(dwconv1(v_inp))  (into k_inp buf)
    dwconv3x3<<<BATCH * HGT * WID, 256, 0, stream>>>(v_inp, c1w, k_inp, 1, 0);

    // 6) xo = attn @ (v*mask)  (WMMA, into q_inp buf)
    attn_apply_v<<<4096, 256, 0, stream>>>(v_inp, mask, attn, q_inp);

    // 7) out_c = xo @ Wp + bp  -> d_out
    gemm_f32_wmma<<<gemmBlocks, 256, 0, stream>>>(q_inp, Wp, bp, out, MROWS, CH, CH);

    // 8) d_out += dwconv2(tmp)
    dwconv3x3<<<BATCH * HGT * WID, 256, 0, stream>>>(k_inp, c2w, out, 0, 1);
}